// VectorQuantizerEMA_44435731644781
// MI455X (gfx1250) — compile-verified
//
#include <hip/hip_runtime.h>

typedef __attribute__((ext_vector_type(2))) float v2f;
typedef __attribute__((ext_vector_type(4))) float v4f;
typedef __attribute__((ext_vector_type(8))) float v8f;
typedef __attribute__((ext_vector_type(4))) unsigned int v4u;
typedef __attribute__((ext_vector_type(4))) int v4i;
typedef __attribute__((ext_vector_type(8))) int v8i;

#define N_ROWS 32768
#define K_CODES 4096
#define D_DIM 256
#define ZS 260          // padded LDS row stride (dwords) -> conflict-free WMMA operand reads
#define ROWS_PER_BLK 128
#define WAVES 8
#define NTILES (K_CODES / 16)

// dynamic-LDS carve (dwords): es at 0 so B-operand addresses fit DS immediates
#define ES_OFF   0
#define ENS_OFF  (2 * 16 * ZS)            // 8320
#define ZSV_OFF  (ENS_OFF + K_CODES)      // 12416
#define SMEM_DW  (ZSV_OFF + ROWS_PER_BLK * ZS)
#define SMEM_BYTES ((size_t)SMEM_DW * 4)  // 182,784 B dynamic (+ ~17 KB static)

#if defined(__has_builtin)
#if __has_builtin(__builtin_amdgcn_tensor_load_to_lds)
#define HAVE_TDM 1
#endif
#if __has_builtin(__builtin_amdgcn_s_wait_tensorcnt)
#define HAVE_TCNT 1
#endif
#endif

// ---------------------------------------------------------------------------
// Kernel 1: e_norm[k] = sum_d embed[k][d]^2
// ---------------------------------------------------------------------------
__global__ __launch_bounds__(256) void vq_enorm_kernel(
    const float* __restrict__ embed, float* __restrict__ enorm) {
  int k = blockIdx.x * blockDim.x + threadIdx.x;
  if (k >= K_CODES) return;
  const float* row = embed + (size_t)k * D_DIM;
  float s = 0.f;
#pragma unroll 8
  for (int d = 0; d < D_DIM; d += 4) {
    v4f v = *(const v4f*)(row + d);
    s += v.x * v.x + v.y * v.y + v.z * v.z + v.w * v.w;
  }
  enorm[k] = s;
}

// ---------------------------------------------------------------------------
// Kernel 2: zero-fill the one_hot region (float4 stores, grid-stride)
// ---------------------------------------------------------------------------
__global__ __launch_bounds__(256) void vq_zero_kernel(
    float* __restrict__ p, size_t n4) {
  size_t i = (size_t)blockIdx.x * blockDim.x + threadIdx.x;
  size_t stride = (size_t)gridDim.x * blockDim.x;
  v4f z = {0.f, 0.f, 0.f, 0.f};
  for (; i < n4; i += stride) ((v4f*)p)[i] = z;
}

// ---------------------------------------------------------------------------
// TDM: load one 16x256 f32 embed tile into LDS with a 4-dword pad after every
// 256 dwords (-> 260-dword row stride).  D# packed per CDNA5 ISA
// sections 8.3/8.4.
// ---------------------------------------------------------------------------
#if defined(HAVE_TDM)
__device__ __forceinline__ void tdm_load_tile(const float* gsrc, unsigned lds_off) {
  unsigned long long ga = (unsigned long long)gsrc;
  v4u g0;
  g0[0] = 1u;                                        // count=1, user mode
  g0[1] = lds_off;                                   // lds_addr (bytes)
  g0[2] = (unsigned)(ga & 0xffffffffu);              // global_addr[31:0]
  g0[3] = (unsigned)((ga >> 32) & 0x01ffffffu)       // global_addr[56:32]
          | (2u << 30);                              // type=2 ("image")
  v8i g1;
  g1[0] = (int)((2u << 16)        // data_size = 4 bytes
              | (1u << 20)        // pad_enable
              | (7u << 22)        // pad_interval = 256 dwords
              | (3u << 25));      // pad_amount   = 4 dwords
  g1[1] = (int)(256u << 16);      // tensor_dim0[15:0] = 256 (bits 63:48)
  g1[2] = (int)(16u << 16);       // tensor_dim0 hi=0 | tensor_dim1 lo = 16
  g1[3] = (int)(256u << 16);      // tensor_dim1 hi=0 | tile_dim0 = 256
  g1[4] = 16;                     // tile_dim1 = 16, tile_dim2 = 0
  g1[5] = 256;                    // tensor_dim0_stride[31:0] = 256
  g1[6] = 0;
  g1[7] = 0;
  v4i gz = {0, 0, 0, 0};
#if __clang_major__ >= 23
  v8i gz8 = {0, 0, 0, 0, 0, 0, 0, 0};
  __builtin_amdgcn_tensor_load_to_lds(g0, g1, gz, gz, gz8, 0);
#else
  __builtin_amdgcn_tensor_load_to_lds(g0, g1, gz, gz, 0);
#endif
}
#endif

__device__ __forceinline__ void wait_tensorcnt0() {
#if defined(HAVE_TCNT)
  __builtin_amdgcn_s_wait_tensorcnt(0);
#else
  asm volatile("s_wait_tensorcnt 0x0" ::: "memory");
#endif
}

// ---------------------------------------------------------------------------
// Kernel 3: 128 rows per workgroup (8 waves x 16 rows).  Embed tiles are
// TDM-staged into a double-buffered padded LDS tile shared by all 8 waves.
// Dynamic-LDS carving pins es/ens at low offsets (B-loads get 16-bit DS
// immediates / 2addr merging); the 130 KB z tile sits above them and is
// addressed from a once-computed per-wave base register.
// ---------------------------------------------------------------------------
__global__ __launch_bounds__(256) void vq_main_kernel(
    const float* __restrict__ z_e, const float* __restrict__ embed,
    const float* __restrict__ enorm,
    float* __restrict__ zq, float* __restrict__ idx_out,
    float* __restrict__ onehot) {
  extern __shared__ __align__(16) float smem[];
  float* es  = smem + ES_OFF;    // [2][16*ZS]  double-buffered B tiles
  float* ens = smem + ENS_OFF;   // [K_CODES]   ||e||^2
  float* zs  = smem + ZSV_OFF;   // [128*ZS]    z tile
  __shared__ float rbest[WAVES][16][16];
  __shared__ int   ridx[WAVES][16][16];
  __shared__ int   idx16[WAVES][16];

  const int tid = threadIdx.x;       // 0..255
  const int lane = tid & 31;
  const int wid = tid >> 5;          // wave id 0..7
  const int lm = lane & 15;
  const int half = lane >> 4;
  const int blk_row0 = blockIdx.x * ROWS_PER_BLK;
  const int row0 = blk_row0 + wid * 16;   // this wave's 16 rows

  // ---- stage z tile (128 x 256 f32) into LDS, coalesced float4 -------------
#pragma unroll
  for (int i = 0; i < 32; ++i) {
    int e = tid + 256 * i;           // float4 index 0..8191
    int m = e >> 6;                  // 64 float4 per row
    int dd = (e & 63) << 2;
    v4f v = *(const v4f*)(z_e + (size_t)(blk_row0 + m) * D_DIM + dd);
    *(v4f*)(&zs[m * ZS + dd]) = v;
  }
  // ---- stage ||e||^2 into LDS ---------------------------------------------
#pragma unroll
  for (int i = 0; i < 4; ++i) {
    int e = (tid + 256 * i) << 2;
    *(v4f*)(&ens[e]) = *(const v4f*)(enorm + e);
  }

  // ---- prologue: TDM prefetch of tile 0 (wave 0 only; per-wave DMA op) -----
#if defined(HAVE_TDM)
  if (tid < 32) {
    tdm_load_tile(embed, (unsigned)(size_t)&es[0]);
  }
#endif

  float best[8];
  int   bidx[8];
#pragma unroll
  for (int r = 0; r < 8; ++r) { best[r] = __builtin_inff(); bidx[r] = 0; }

  // A layout (32-bit 16x4): VGPR j, lane l -> K = 2*(l>=16)+j, M = l&15.
  const int dbase = 2 * half;
  const int zoff = (wid * 16 + lm) * ZS + dbase;
  const int boff0 = lm * ZS + dbase;

  for (int t = 0; t < NTILES; ++t) {
#if defined(HAVE_TDM)
    if (tid < 32) wait_tensorcnt0();      // tile t landed in buffer t&1
    __syncthreads();                      // everyone sees it; prior buffer free
    if (tid < 32 && (t + 1) < NTILES) {   // overlap DMA of t+1 with compute of t
      tdm_load_tile(embed + (size_t)(t + 1) * 16 * D_DIM,
                    (unsigned)(size_t)&es[((t + 1) & 1) * 16 * ZS]);
    }
#else
    __syncthreads();                      // previous users done with buffer
    {
      // synchronous cooperative stage of tile t: 1024 float4 by 256 threads
      const float* src = embed + (size_t)t * 16 * D_DIM;
      float* dstb = es + (t & 1) * 16 * ZS;
      int m = tid >> 4;                   // 16 float4 per row-segment
      int dd = (tid & 15) << 4;
      v4f v0 = *(const v4f*)(src + m * D_DIM + dd);
      v4f v1 = *(const v4f*)(src + m * D_DIM + dd + 4);
      v4f v2 = *(const v4f*)(src + m * D_DIM + dd + 8);
      v4f v3 = *(const v4f*)(src + m * D_DIM + dd + 12);
      *(v4f*)(&dstb[m * ZS + dd]) = v0;
      *(v4f*)(&dstb[m * ZS + dd + 4]) = v1;
      *(v4f*)(&dstb[m * ZS + dd + 8]) = v2;
      *(v4f*)(&dstb[m * ZS + dd + 12]) = v3;
    }
    __syncthreads();
#endif

    const int nb = t * 16 + lm;
    const float en = ens[nb];
    const float* bbuf = es + (t & 1) * 16 * ZS + boff0;

    v8f acc = {};
#pragma unroll 8
    for (int d0 = 0; d0 < D_DIM; d0 += 4) {
      v2f a = *(const v2f*)(&zs[zoff + d0]);   // ds_load_2addr_b64 (A)
      v2f b = *(const v2f*)(&bbuf[d0]);        // ds_load_2addr_b64 (B, low LDS)
      acc = __builtin_amdgcn_wmma_f32_16x16x4_f32(
          false, a, false, b, (short)0, acc, false, false);
    }
    // C/D layout: lane l, VGPR r -> (M = r + 8*(l>=16), N = t*16 + (l&15))
#pragma unroll
    for (int r = 0; r < 8; ++r) {
      float v = en - 2.0f * acc[r];
      if (v < best[r]) { best[r] = v; bidx[r] = nb; }
    }
  }

  // ---- cross-lane argmin reduction per row --------------------------------
#pragma unroll
  for (int r = 0; r < 8; ++r) {
    int mrow = r + 8 * half;
    rbest[wid][mrow][lm] = best[r];
    ridx[wid][mrow][lm]  = bidx[r];
  }
  __syncthreads();

  if (lane < 16) {
    float m = rbest[wid][lane][0];
    int mi = ridx[wid][lane][0];
    for (int j = 1; j < 16; ++j) {
      float v = rbest[wid][lane][j];
      int vi = ridx[wid][lane][j];
      if (v < m || (v == m && vi < mi)) { m = v; mi = vi; }  // first-index tie-break
    }
    idx16[wid][lane] = mi;
    idx_out[row0 + lane] = (float)mi;
    onehot[(size_t)(row0 + lane) * K_CODES + mi] = 1.0f;     // after zero-fill
  }
  __syncthreads();

  // ---- z_q = embed[idx] gather: each wave its 16 rows ---------------------
  for (int m = 0; m < 16; ++m) {
    const float* src = embed + (size_t)idx16[wid][m] * D_DIM;
    float* dst = zq + (size_t)(row0 + m) * D_DIM;
#pragma unroll
    for (int i = 0; i < 2; ++i) {
      int off = (lane + 32 * i) << 2;
      *(v4f*)(dst + off) = *(const v4f*)(src + off);
    }
  }
}

// ---------------------------------------------------------------------------
// Launch: d_out layout (all f32) = [ z_q (N*D) | idx-as-float (N) | one_hot (N*K) ]
// ---------------------------------------------------------------------------
extern "C" void kernel_launch(void* const* d_in, const int* in_sizes, int n_in,
                              void* d_out, int out_size, void* d_ws, size_t ws_size,
                              hipStream_t stream) {
  (void)in_sizes; (void)n_in; (void)out_size; (void)ws_size;
  const float* z_e   = (const float*)d_in[0];
  const float* embed = (const float*)d_in[1];
  float* out    = (float*)d_out;
  float* zq     = out;
  float* idxf   = out + (size_t)N_ROWS * D_DIM;
  float* onehot = idxf + N_ROWS;
  float* enorm  = (float*)d_ws;   // K floats of scratch

  vq_enorm_kernel<<<K_CODES / 256, 256, 0, stream>>>(embed, enorm);

  size_t n4 = (size_t)N_ROWS * K_CODES / 4;
  vq_zero_kernel<<<8192, 256, 0, stream>>>(onehot, n4);

  vq_main_kernel<<<N_ROWS / ROWS_PER_BLK, 256, SMEM_BYTES, stream>>>(
      z_e, embed, enorm, zq, idxf, onehot);
}